// GCN_23287312679643
// MI455X (gfx1250) — compile-verified
//
#include <hip/hip_runtime.h>

#define HID 128

typedef __attribute__((ext_vector_type(2))) float v2f;
typedef __attribute__((ext_vector_type(8))) float v8f;

// ---------------- small helpers ----------------

__global__ void zero_int_kernel(int* __restrict__ p, int n) {
    int i = blockIdx.x * blockDim.x + threadIdx.x;
    if (i < n) p[i] = 0;
}

__global__ void fill_f32_kernel(float* __restrict__ p, float v, int n) {
    int i = blockIdx.x * blockDim.x + threadIdx.x;
    if (i < n) p[i] = v;
}

// in-degree count over real edges (self-loops handled analytically later)
__global__ void deg_count_kernel(const int* __restrict__ dst, int* __restrict__ deg, int E) {
    int i = blockIdx.x * blockDim.x + threadIdx.x;
    if (i < E) atomicAdd(&deg[dst[i]], 1);
}

// dinv[i] = rsqrt(deg_in[i] + 1)   (+1 = self loop, matches reference)
__global__ void dinv_kernel(const int* __restrict__ deg, float* __restrict__ dinv, int n) {
    int i = blockIdx.x * blockDim.x + threadIdx.x;
    if (i < n) dinv[i] = rsqrtf((float)deg[i] + 1.0f);
}

// CSR slot allocation: block-level exclusive scan in LDS + one global atomic per block.
__global__ __launch_bounds__(256) void csr_alloc_kernel(const int* __restrict__ deg,
                                                        int* __restrict__ start,
                                                        int* __restrict__ wcur,
                                                        int* __restrict__ counter, int N) {
    __shared__ int sh[256];
    __shared__ int base;
    const int tid = threadIdx.x;
    const int i = blockIdx.x * 256 + tid;
    int v = (i < N) ? deg[i] : 0;
    sh[tid] = v;
    __syncthreads();
    // inclusive Hillis-Steele scan
    for (int o = 1; o < 256; o <<= 1) {
        int t = (tid >= o) ? sh[tid - o] : 0;
        __syncthreads();
        sh[tid] += t;
        __syncthreads();
    }
    if (tid == 255) base = atomicAdd(counter, sh[255]);
    __syncthreads();
    if (i < N) {
        int s = base + (sh[tid] - v);   // exclusive prefix within block + block base
        start[i] = s;
        wcur[i]  = s;
    }
}

// place each edge's src into its dst's CSR slot range
__global__ void csr_fill_kernel(const int* __restrict__ src, const int* __restrict__ dst,
                                int* __restrict__ wcur, int* __restrict__ col, int E) {
    int i = blockIdx.x * blockDim.x + threadIdx.x;
    if (i < E) {
        int slot = atomicAdd(&wcur[dst[i]], 1);
        col[slot] = src[i];
    }
}

// ---------------- dense GEMM: G[N,128] = (X[N,128] @ W[128,128]) * dinv[row]  (fp32 WMMA) ----------------
// Block = 256 threads = 8 waves; block handles 16 rows, wave w handles cols [16w,16w+16).
// Fragment addressing (ISA 7.12.2, f32 16x16x4):
//   A: M = lane&15, elem e -> K = kbase + (lane>>4)*2 + e
//   B: N = lane&15, elem e -> K = kbase + (lane>>4)*2 + e
//   C/D: VGPR r -> M = r + (lane>>4)*8, N = lane&15
__global__ __launch_bounds__(256) void gemm128_wmma_scaled(const float* __restrict__ X,
                                                           const float* __restrict__ W,
                                                           const float* __restrict__ dinv,
                                                           float* __restrict__ G, int N) {
    const int lane  = threadIdx.x & 31;
    const int wave  = threadIdx.x >> 5;
    const int khalf = lane >> 4;
    const int l15   = lane & 15;
    const int row_base = blockIdx.x * 16;
    const int col = wave * 16 + l15;

    int arow = row_base + l15;
    if (arow >= N) arow = N - 1;          // clamp; EXEC stays all-1s through WMMA
    const float* __restrict__ xrow = X + (size_t)arow * HID;

    v8f acc = {};
#pragma unroll
    for (int k = 0; k < HID; k += 4) {
        const int ka = k + khalf * 2;
        v2f a, b;
        a.x = xrow[ka];
        a.y = xrow[ka + 1];
        b.x = W[(size_t)ka * HID + col];
        b.y = W[(size_t)(ka + 1) * HID + col];
        acc = __builtin_amdgcn_wmma_f32_16x16x4_f32(false, a, false, b,
                                                    (short)0, acc, false, false);
    }

#pragma unroll
    for (int r = 0; r < 8; ++r) {
        const int orow = row_base + khalf * 8 + r;
        if (orow < N) G[(size_t)orow * HID + col] = acc[r] * dinv[orow];
    }
}

// ---------------- pull aggregation, fused self-loop + bias + ReLU ----------------
// out[d] = relu( dinv[d] * ( g[d] + sum_{s in in(d)} g[s] ) + b )
// One wave per node; lane owns float4 (32*4 = 128 floats).
__global__ __launch_bounds__(256) void pull_kernel(const float* __restrict__ g,
                                                   const int* __restrict__ col,
                                                   const int* __restrict__ start,
                                                   const int* __restrict__ deg,
                                                   const float* __restrict__ dinv,
                                                   const float* __restrict__ bias,
                                                   float* __restrict__ out, int N) {
    long gid = (long)blockIdx.x * blockDim.x + threadIdx.x;
    int node = (int)(gid >> 5);
    int lane = threadIdx.x & 31;
    if (node >= N) return;

    const int p0  = start[node];
    const int cnt = deg[node];

    float4 acc = ((const float4*)(g + (size_t)node * HID))[lane];   // self loop term
    for (int i = 0; i < cnt; ++i) {
        int s = col[p0 + i];                                        // wave-uniform load
        float4 v = ((const float4*)(g + (size_t)s * HID))[lane];
        acc.x += v.x; acc.y += v.y; acc.z += v.z; acc.w += v.w;
    }
    const float dd = dinv[node];
    const float4 bb = ((const float4*)bias)[lane];
    float4 r;
    r.x = fmaxf(acc.x * dd + bb.x, 0.0f);
    r.y = fmaxf(acc.y * dd + bb.y, 0.0f);
    r.z = fmaxf(acc.z * dd + bb.z, 0.0f);
    r.w = fmaxf(acc.w * dd + bb.w, 0.0f);
    ((float4*)(out + (size_t)node * HID))[lane] = r;
}

// ---------------- pooling + FC head ----------------
__global__ __launch_bounds__(128) void colsum_kernel(const float* __restrict__ H,
                                                     float* __restrict__ pooled,
                                                     int N, int rowsPerBlock) {
    const int col = threadIdx.x;          // blockDim.x == 128 -> coalesced
    int r0 = blockIdx.x * rowsPerBlock;
    int r1 = r0 + rowsPerBlock;
    if (r1 > N) r1 = N;
    float s = 0.0f;
    for (int r = r0; r < r1; ++r) s += H[(size_t)r * HID + col];
    atomicAdd(&pooled[col], s);
}

__global__ void final_fc_kernel(const float* __restrict__ pooled,
                                const float* __restrict__ fc_w,
                                const float* __restrict__ fc_b,
                                float* __restrict__ out, float invN, int outdim) {
    int j = threadIdx.x;
    if (j < outdim) {
        float s = fc_b[j];
        for (int k = 0; k < HID; ++k) s += pooled[k] * invN * fc_w[k * outdim + j];
        out[j] = s;
    }
}

// ---------------- driver ----------------
extern "C" void kernel_launch(void* const* d_in, const int* in_sizes, int n_in,
                              void* d_out, int out_size, void* d_ws, size_t ws_size,
                              hipStream_t stream) {
    const float* x   = (const float*)d_in[0];
    const int*   ei  = (const int*)d_in[1];   // [2,E]: row 0 = src, row 1 = dst
    const float* W1  = (const float*)d_in[2];
    const float* b1  = (const float*)d_in[3];
    const float* W2  = (const float*)d_in[4];
    const float* b2  = (const float*)d_in[5];
    const float* W3  = (const float*)d_in[6];
    const float* b3  = (const float*)d_in[7];
    const float* fcw = (const float*)d_in[8];
    const float* fcb = (const float*)d_in[9];
    float* out = (float*)d_out;

    const int N    = in_sizes[0] / HID;
    const int E    = in_sizes[1] / 2;
    const int OUTD = in_sizes[9];
    const int* srcE = ei;
    const int* dstE = ei + E;

    // workspace layout
    float* buf0    = (float*)d_ws;                    // layer output        [N*HID]
    float* buf1    = buf0 + (size_t)N * HID;          // g = scaled GEMM out [N*HID]
    float* dinv    = buf1 + (size_t)N * HID;          // [N]
    float* pooled  = dinv + N;                        // [HID]
    int*   deg     = (int*)(pooled + HID);            // [N]
    int*   start   = deg + N;                         // [N]
    int*   wcur    = start + N;                       // [N]
    int*   counter = wcur + N;                        // [1]
    int*   col     = counter + 1;                     // [E]

    const int T = 256;
    const int nBlkN = (N + T - 1) / T;
    const int nBlkE = (E + T - 1) / T;

    // ---- build normalization + CSR (once per call; int atomics only) ----
    zero_int_kernel<<<nBlkN, T, 0, stream>>>(deg, N);
    deg_count_kernel<<<nBlkE, T, 0, stream>>>(dstE, deg, E);
    dinv_kernel<<<nBlkN, T, 0, stream>>>(deg, dinv, N);
    zero_int_kernel<<<1, 32, 0, stream>>>(counter, 1);
    csr_alloc_kernel<<<nBlkN, T, 0, stream>>>(deg, start, wcur, counter, N);
    csr_fill_kernel<<<nBlkE, T, 0, stream>>>(srcE, dstE, wcur, col, E);

    // ---- 3 GCN layers: WMMA GEMM (row-scaled) + fused pull/bias/ReLU ----
    const int gemmBlocks = (N + 15) / 16;
    const int pullBlocks = (int)(((long)N * 32 + T - 1) / T);
    const float* in = x;
    const float* Ws[3] = {W1, W2, W3};
    const float* bs[3] = {b1, b2, b3};
    for (int layer = 0; layer < 3; ++layer) {
        gemm128_wmma_scaled<<<gemmBlocks, 256, 0, stream>>>(in, Ws[layer], dinv, buf1, N);
        pull_kernel<<<pullBlocks, T, 0, stream>>>(buf1, col, start, deg, dinv, bs[layer], buf0, N);
        in = buf0;
    }

    // ---- global mean pool + FC ----
    fill_f32_kernel<<<1, 128, 0, stream>>>(pooled, 0.0f, HID);
    const int rowsPerBlock = 512;
    colsum_kernel<<<(N + rowsPerBlock - 1) / rowsPerBlock, 128, 0, stream>>>(buf0, pooled, N, rowsPerBlock);
    final_fc_kernel<<<1, 64, 0, stream>>>(pooled, fcw, fcb, out, 1.0f / (float)N, OUTD);
}